// MixtralMoE_85341000171521
// MI455X (gfx1250) — compile-verified
//
#include <hip/hip_runtime.h>
#include <stdint.h>

#define T_TOK 8192
#define H_DIM 2048
#define I_DIM 7168
#define E_NUM 8

#define MTILE   32            // tokens per workgroup tile
#define THREADS 512           // 16 waves (wave32)
#define NWAVE   (THREADS / 32)
#define ICHUNK  256           // i-columns produced per outer iteration (NWAVE*16)
#define HSTR    260           // hlds row stride in floats (bank-conflict-free b64 reads)

typedef __attribute__((ext_vector_type(2))) float v2f;
typedef __attribute__((ext_vector_type(8))) float v8f;

// D = A(16x4 f32) * B(4x16 f32) + C(16x16 f32), wave32 WMMA.
__device__ __forceinline__ v8f wmma_f32(v2f a, v2f b, v8f c) {
  return __builtin_amdgcn_wmma_f32_16x16x4_f32(false, a, false, b, (short)0, c,
                                               false, false);
}

// ---------------------------------------------------------------------------
// Router: one wave per token. logits = x . gate_w^T, softmax, top-2, renorm,
// atomic-append (token, weight) into per-expert compacted lists.
// ---------------------------------------------------------------------------
__global__ void router_kernel(const float* __restrict__ x,
                              const float* __restrict__ gw,
                              int* __restrict__ counts,
                              int* __restrict__ tok_list,
                              float* __restrict__ w_list) {
  const int lane = threadIdx.x & 31;
  const int wid  = threadIdx.x >> 5;
  const int t = blockIdx.x * (blockDim.x >> 5) + wid;
  if (t >= T_TOK) return;

  float acc[E_NUM];
#pragma unroll
  for (int e = 0; e < E_NUM; ++e) acc[e] = 0.f;

  const float* xr = x + (size_t)t * H_DIM;
  for (int k = lane; k < H_DIM; k += 32) {
    const float xv = xr[k];
#pragma unroll
    for (int e = 0; e < E_NUM; ++e)
      acc[e] = fmaf(xv, gw[(size_t)e * H_DIM + k], acc[e]);
  }
#pragma unroll
  for (int e = 0; e < E_NUM; ++e) {
    for (int off = 16; off > 0; off >>= 1)
      acc[e] += __shfl_xor(acc[e], off, 32);
  }

  if (lane == 0) {
    float mx = acc[0];
#pragma unroll
    for (int e = 1; e < E_NUM; ++e) mx = fmaxf(mx, acc[e]);
    float p[E_NUM];
#pragma unroll
    for (int e = 0; e < E_NUM; ++e) p[e] = __expf(acc[e] - mx);
    // top-2 (lowest index wins ties, matching lax.top_k)
    int i0 = 0; float v0 = p[0];
#pragma unroll
    for (int e = 1; e < E_NUM; ++e) if (p[e] > v0) { v0 = p[e]; i0 = e; }
    int i1 = (i0 == 0) ? 1 : 0; float v1 = p[i1];
#pragma unroll
    for (int e = 0; e < E_NUM; ++e)
      if (e != i0 && p[e] > v1) { v1 = p[e]; i1 = e; }
    const float inv = 1.f / (v0 + v1);

    int s0 = atomicAdd(&counts[i0], 1);
    tok_list[(size_t)i0 * T_TOK + s0] = t;
    w_list [(size_t)i0 * T_TOK + s0] = v0 * inv;
    int s1 = atomicAdd(&counts[i1], 1);
    tok_list[(size_t)i1 * T_TOK + s1] = t;
    w_list [(size_t)i1 * T_TOK + s1] = v1 * inv;
  }
}

// ---------------------------------------------------------------------------
// Expert kernel: grid (E, T/MTILE). Per 32-token tile (gathered rows):
//   Phase A: h[32, 256-chunk] = silu(x w1^T) * (x w3^T)  (fp32 WMMA, K=2048)
//   Phase B: out_acc[32, 2048] += h_chunk @ w2_chunk^T    (fp32 WMMA)
// Epilogue: scale by routing weight, atomic-add into out (2 adds/element).
// ---------------------------------------------------------------------------
__global__ __launch_bounds__(THREADS, 1)
void moe_expert_kernel(const float* __restrict__ x,
                       const float* __restrict__ w1,
                       const float* __restrict__ w3,
                       const float* __restrict__ w2,
                       const int* __restrict__ counts,
                       const int* __restrict__ tok_list,
                       const float* __restrict__ w_list,
                       float* __restrict__ out) {
  __shared__ float hlds[MTILE * HSTR];   // 32*260*4 = 33280 B
  __shared__ int   tok_s[MTILE];
  __shared__ float wt_s[MTILE];

  const int e     = blockIdx.x;
  const int cnt   = counts[e];
  const int tile0 = blockIdx.y * MTILE;
  if (tile0 >= cnt) return;              // uniform early-exit

  const int tid   = threadIdx.x;
  const int lane  = tid & 31;
  const int wid   = tid >> 5;            // 0..15
  const int l15   = lane & 15;
  const int khalf = lane >> 4;           // 0 or 1
  const int koff  = khalf * 2;           // K offset inside 4-wide step (A & B frags)
  const int moff  = khalf * 8;           // C/D row offset for upper lane half

  if (tid < MTILE) {
    const int idx = tile0 + tid;
    const bool v  = idx < cnt;
    tok_s[tid] = tok_list[(size_t)e * T_TOK + (v ? idx : tile0)];
    wt_s[tid]  = v ? w_list[(size_t)e * T_TOK + idx] : 0.f;
  }
  __syncthreads();

  const float* xr0 = x + (size_t)tok_s[l15]      * H_DIM;  // A rows, m-tile 0
  const float* xr1 = x + (size_t)tok_s[16 + l15] * H_DIM;  // A rows, m-tile 1

  const float* w1e = w1 + (size_t)e * I_DIM * H_DIM;
  const float* w3e = w3 + (size_t)e * I_DIM * H_DIM;
  const float* w2e = w2 + (size_t)e * H_DIM * I_DIM;

  const v8f zero8 = {0.f, 0.f, 0.f, 0.f, 0.f, 0.f, 0.f, 0.f};
  v8f accO[2][8];                        // 32 rows x 128 cols per wave
#pragma unroll
  for (int mt = 0; mt < 2; ++mt)
#pragma unroll
    for (int nt = 0; nt < 8; ++nt) accO[mt][nt] = zero8;

  const int nbase = wid * 128;           // this wave's output-column slice

  for (int i0 = 0; i0 < I_DIM; i0 += ICHUNK) {
    // ---------------- Phase A: this wave's 16 i-columns ----------------
    const float* r1 = w1e + (size_t)(i0 + wid * 16 + l15) * H_DIM;  // B col n=l15
    const float* r3 = w3e + (size_t)(i0 + wid * 16 + l15) * H_DIM;
    v8f a1m0 = zero8, a1m1 = zero8, a3m0 = zero8, a3m1 = zero8;
#pragma unroll 4
    for (int ks = 0; ks < H_DIM; ks += 4) {
      const int ka = ks + koff;
      const v2f A0 = *(const v2f*)(xr0 + ka);
      const v2f A1 = *(const v2f*)(xr1 + ka);
      const v2f B1 = *(const v2f*)(r1 + ka);
      const v2f B3 = *(const v2f*)(r3 + ka);
      a1m0 = wmma_f32(A0, B1, a1m0);
      a1m1 = wmma_f32(A1, B1, a1m1);
      a3m0 = wmma_f32(A0, B3, a3m0);
      a3m1 = wmma_f32(A1, B3, a3m1);
    }
    __syncthreads();                     // previous Phase B done reading hlds
    {
      const int col = wid * 16 + l15;
#pragma unroll
      for (int r = 0; r < 8; ++r) {
        const float g0 = a1m0[r];
        const float g1 = a1m1[r];
        const float h0 = (g0 / (1.f + __expf(-g0))) * a3m0[r];
        const float h1 = (g1 / (1.f + __expf(-g1))) * a3m1[r];
        hlds[(r + moff) * HSTR + col]        = h0;   // m-tile 0: rows 0..15
        hlds[(16 + r + moff) * HSTR + col]   = h1;   // m-tile 1: rows 16..31
      }
    }
    __syncthreads();
    // ---------------- Phase B: accumulate out columns ----------------
#pragma unroll 2
    for (int ks = 0; ks < ICHUNK; ks += 4) {
      const v2f HA0 = *(const v2f*)(&hlds[l15 * HSTR + ks + koff]);
      const v2f HA1 = *(const v2f*)(&hlds[(16 + l15) * HSTR + ks + koff]);
      const size_t kg = (size_t)i0 + ks + koff;
#pragma unroll
      for (int nt = 0; nt < 8; ++nt) {
        const float* r2 = w2e + (size_t)(nbase + nt * 16 + l15) * I_DIM + kg;
        const v2f B2 = *(const v2f*)r2;
        accO[0][nt] = wmma_f32(HA0, B2, accO[0][nt]);
        accO[1][nt] = wmma_f32(HA1, B2, accO[1][nt]);
      }
    }
  }

  // ---------------- epilogue: scale + scatter-add ----------------
#pragma unroll
  for (int mt = 0; mt < 2; ++mt) {
#pragma unroll
    for (int r = 0; r < 8; ++r) {
      const int m = mt * 16 + r + moff;
      if (tile0 + m < cnt) {
        const float wgt = wt_s[m];
        float* orow = out + (size_t)tok_s[m] * H_DIM;
#pragma unroll
        for (int nt = 0; nt < 8; ++nt) {
          const float val = wgt * accO[mt][nt][r];
          __hip_atomic_fetch_add(orow + (nbase + nt * 16 + l15), val,
                                 __ATOMIC_RELAXED, __HIP_MEMORY_SCOPE_AGENT);
        }
      }
    }
  }
}

// ---------------------------------------------------------------------------
extern "C" void kernel_launch(void* const* d_in, const int* in_sizes, int n_in,
                              void* d_out, int out_size, void* d_ws, size_t ws_size,
                              hipStream_t stream) {
  (void)in_sizes; (void)n_in; (void)out_size; (void)ws_size;
  const float* x  = (const float*)d_in[0];
  const float* gw = (const float*)d_in[1];
  const float* w1 = (const float*)d_in[2];
  const float* w3 = (const float*)d_in[3];
  const float* w2 = (const float*)d_in[4];
  float* out = (float*)d_out;

  char* ws = (char*)d_ws;
  int*   counts   = (int*)ws;                                   // 256 B reserved
  int*   tok_list = (int*)(ws + 256);                           // E*T ints (256 KB)
  float* w_list   = (float*)(ws + 256 + (size_t)E_NUM * T_TOK * 4); // 256 KB

  hipMemsetAsync(counts, 0, 256, stream);
  hipMemsetAsync(d_out, 0, (size_t)T_TOK * H_DIM * sizeof(float), stream);

  router_kernel<<<T_TOK / 8, 256, 0, stream>>>(x, gw, counts, tok_list, w_list);

  dim3 grid(E_NUM, T_TOK / MTILE);
  moe_expert_kernel<<<grid, THREADS, 0, stream>>>(x, w1, w3, w2,
                                                  counts, tok_list, w_list, out);
}